// Net_90125593739636
// MI455X (gfx1250) — compile-verified
//
#include <hip/hip_runtime.h>
#include <hip/hip_bf16.h>
#include <math.h>

// ---------------- problem constants (fixed by reference) ----------------
#define NFEAT 100      // features per node (25 float4)
#define NF4   25
#define NCLS  40       // classes
#define NFRAG 75       // 25 k-steps * 3 column tiles of WMMA B fragments

typedef __attribute__((ext_vector_type(2))) float v2f;
typedef __attribute__((ext_vector_type(8))) float v8f;

// lane-uniform broadcast via v_readlane_b32 (no LDS round-trip)
__device__ __forceinline__ float readlane_f(float x, int l) {
    return __int_as_float(__builtin_amdgcn_readlane(__float_as_int(x), l));
}

// xor-shuffle via ds_swizzle (group-of-32: src = lane ^ XM), immediate pattern
template <int XM>
__device__ __forceinline__ float swz_xor(float x) {
    return __int_as_float(
        __builtin_amdgcn_ds_swizzle(__float_as_int(x), (XM << 10) | 0x1f));
}

// ================= CSR build (by destination node) =================
__global__ void k_zero_cnt(int* __restrict__ cnt, int n) {
    int i = blockIdx.x * blockDim.x + threadIdx.x;
    if (i < n) cnt[i] = 0;
}

__global__ void k_count(const int* __restrict__ col, int* __restrict__ cnt, int e) {
    int i = blockIdx.x * blockDim.x + threadIdx.x;
    if (i < e) atomicAdd(&cnt[col[i]], 1);
}

// dinv[n] = 1/sqrt(in_degree + 1)   (self-loop included)
__global__ void k_dinv(const int* __restrict__ cnt, float* __restrict__ dinv, int n) {
    int i = blockIdx.x * blockDim.x + threadIdx.x;
    if (i < n) dinv[i] = rsqrtf((float)cnt[i] + 1.0f);
}

// Single-block exclusive scan: off[i] = sum(cnt[0..i-1]); also seeds pos[] = off[].
__global__ __launch_bounds__(1024) void k_scan(const int* __restrict__ cnt,
                                               int* __restrict__ off,
                                               int* __restrict__ pos, int n) {
    __shared__ int wsum[32];
    __shared__ int chunk_total;
    __shared__ int carry;
    int lane = threadIdx.x & 31;
    int wid  = threadIdx.x >> 5;
    if (threadIdx.x == 0) carry = 0;
    __syncthreads();
    for (int base = 0; base < n; base += 1024) {
        int i = base + threadIdx.x;
        int v = (i < n) ? cnt[i] : 0;
        int s = v;                                   // inclusive wave scan
#pragma unroll
        for (int d = 1; d < 32; d <<= 1) {
            int t = __shfl_up(s, (unsigned)d, 32);
            if (lane >= d) s += t;
        }
        if (lane == 31) wsum[wid] = s;
        __syncthreads();
        if (wid == 0) {                              // scan the 32 wave sums
            int wv = wsum[lane];
            int ws = wv;
#pragma unroll
            for (int d = 1; d < 32; d <<= 1) {
                int t = __shfl_up(ws, (unsigned)d, 32);
                if (lane >= d) ws += t;
            }
            if (lane == 31) chunk_total = ws;
            wsum[lane] = ws - wv;                    // exclusive wave offset
        }
        __syncthreads();
        int excl = (s - v) + wsum[wid] + carry;
        if (i < n) { off[i] = excl; pos[i] = excl; }
        __syncthreads();
        if (threadIdx.x == 0) carry += chunk_total;
        __syncthreads();
    }
}

// scatter edge sources into CSR slots (order within a segment is irrelevant)
__global__ void k_fill(const int* __restrict__ row, const int* __restrict__ col,
                       int* __restrict__ pos, int* __restrict__ srcs, int e) {
    int i = blockIdx.x * blockDim.x + threadIdx.x;
    if (i < e) {
        int p = atomicAdd(&pos[col[i]], 1);
        srcs[p] = row[i];
    }
}

// ================= pull-mode propagation (atomic-free) =================
// One wave per destination node. Lanes 0-24 keep the 100-float accumulator in
// registers; neighbor ids/weights are batch-loaded 32 at a time and broadcast
// with v_readlane (src becomes wave-uniform -> scalar-base global loads).
__global__ __launch_bounds__(256) void k_pull(const float* __restrict__ hin,
                                              float* __restrict__ hout,
                                              const int* __restrict__ off,
                                              const int* __restrict__ cnt,
                                              const int* __restrict__ srcs,
                                              const float* __restrict__ dinv, int n) {
    int gw   = (blockIdx.x * blockDim.x + threadIdx.x) >> 5; // wave id = node
    int lane = threadIdx.x & 31;
    if (gw >= n) return;
    int node = gw;
    int s0 = off[node];
    int c  = cnt[node];
    float dn = dinv[node];
    const float4* hin4 = (const float4*)hin;
    float4 acc = make_float4(0.f, 0.f, 0.f, 0.f);
    if (lane < NF4) {                                // self-loop term
        float4 v = hin4[node * NF4 + lane];
        float w = dn * dn;
        acc.x = w * v.x; acc.y = w * v.y; acc.z = w * v.z; acc.w = w * v.w;
    }
    for (int k0 = 0; k0 < c; k0 += 32) {
        int kc = c - k0; if (kc > 32) kc = 32;
        int   sk = 0;
        float wk = 0.f;
        if (lane < kc) {                             // batch-load 32 (src, weight)
            sk = srcs[s0 + k0 + lane];
            wk = dn * dinv[sk];
        }
        for (int k = 0; k < kc; ++k) {
            int   src = __builtin_amdgcn_readlane(sk, k);  // wave-uniform
            float w2  = readlane_f(wk, k);
            if (lane < NF4) {
                float4 u = hin4[src * NF4 + lane];   // scalar base + lane offset
                acc.x += w2 * u.x; acc.y += w2 * u.y;
                acc.z += w2 * u.z; acc.w += w2 * u.w;
            }
        }
    }
    if (lane < NF4) ((float4*)hout)[node * NF4 + lane] = acc;
}

// ================= fused GEMM (f32 WMMA) + bias + log_softmax =================
// 5 waves/block, one 16-row tile per wave (50000 = 625*5*16 exactly).
// K=100 -> 25 x V_WMMA_F32_16X16X4_F32 per column tile; 3 column tiles (40->48).
// W pre-swizzled in LDS into per-lane B fragments (one ds_load_b64 each).
__global__ __launch_bounds__(160) void k_gemm_logsoftmax(
        const float* __restrict__ h, const float* __restrict__ W,
        const float* __restrict__ b, float* __restrict__ out, int nRowTiles) {
    __shared__ v2f WsB[NFRAG * 32];                 // 75 frags * 32 lanes * 8B = 19.2 KB
    for (int i = threadIdx.x; i < NFRAG * 32; i += blockDim.x) {
        int frag = i >> 5, ln = i & 31;
        int kk = frag / 3, j = frag % 3;
        int hf = ln >> 4, mm = ln & 15;
        int k0 = kk * 4 + 2 * hf;                   // B: lanes0-15 K=k0,k0+1; 16-31 K=k0+2,k0+3
        int cj = j * 16 + mm;
        v2f val;
        val.x = (cj < NCLS) ? W[k0 * NCLS + cj] : 0.0f;
        val.y = (cj < NCLS) ? W[(k0 + 1) * NCLS + cj] : 0.0f;
        WsB[i] = val;
    }
    __syncthreads();

    int wave = threadIdx.x >> 5;
    int lane = threadIdx.x & 31;
    int half = lane >> 4;
    int m    = lane & 15;
    int rowTile = blockIdx.x * 5 + wave;
    if (rowTile >= nRowTiles) return;               // wave-uniform (EXEC stays all-1)
    int row0 = rowTile * 16;

    float bias[3];
#pragma unroll
    for (int j = 0; j < 3; ++j) {
        int cj = j * 16 + m;
        bias[j] = (cj < NCLS) ? b[cj] : 0.0f;       // bias baked into C accumulator
    }
    v8f acc[3];
#pragma unroll
    for (int j = 0; j < 3; ++j)
#pragma unroll
        for (int v = 0; v < 8; ++v) acc[j][v] = bias[j];

    // A: lanes0-15 hold K=k0,k0+1 ; lanes16-31 hold K=k0+2,k0+3 (8B-aligned pairs)
    const float* ap = h + (row0 + m) * NFEAT + 2 * half;
    const v2f* wb = WsB + lane;
#pragma unroll 5
    for (int kk = 0; kk < NFEAT / 4; ++kk) {
        v2f a = *(const v2f*)(ap + kk * 4);          // one b64 load, immediate offset
#pragma unroll
        for (int j = 0; j < 3; ++j) {
            v2f bb = wb[(kk * 3 + j) * 32];          // one ds_load_b64, aligned pair
            acc[j] = __builtin_amdgcn_wmma_f32_16x16x4_f32(
                false, a, false, bb, (short)0, acc[j], false, false);
        }
    }

    // D layout: VGPR v -> row M=v (lanes0-15) / M=8+v (lanes16-31), col = j*16+m.
    // Row-wise log-softmax: ds_swizzle xor-reductions within each 16-lane half.
#pragma unroll
    for (int v = 0; v < 8; ++v) {
        int row = row0 + v + 8 * half;
        float l0 = acc[0][v], l1 = acc[1][v], l2 = acc[2][v];
        bool has2 = (m < 8);                        // cols 40..47 are padding
        float mx = fmaxf(l0, l1);
        if (has2) mx = fmaxf(mx, l2);
        mx = fmaxf(mx, swz_xor<1>(mx));
        mx = fmaxf(mx, swz_xor<2>(mx));
        mx = fmaxf(mx, swz_xor<4>(mx));
        mx = fmaxf(mx, swz_xor<8>(mx));
        float s = expf(l0 - mx) + expf(l1 - mx) + (has2 ? expf(l2 - mx) : 0.0f);
        s += swz_xor<1>(s);
        s += swz_xor<2>(s);
        s += swz_xor<4>(s);
        s += swz_xor<8>(s);
        float lse = mx + logf(s);
        out[(size_t)row * NCLS + m]      = l0 - lse;
        out[(size_t)row * NCLS + 16 + m] = l1 - lse;
        if (has2) out[(size_t)row * NCLS + 32 + m] = l2 - lse;
    }
}

// ================= host-side launch =================
extern "C" void kernel_launch(void* const* d_in, const int* in_sizes, int n_in,
                              void* d_out, int out_size, void* d_ws, size_t ws_size,
                              hipStream_t stream) {
    const float* x  = (const float*)d_in[0];
    const int*   ei = (const int*)d_in[1];        // [2, E]: row (src), col (dst)
    const float* W  = (const float*)d_in[2];      // [100, 40]
    const float* b  = (const float*)d_in[3];      // [40]
    float* out = (float*)d_out;

    const int N = in_sizes[0] / NFEAT;            // 50000
    const int E = in_sizes[1] / 2;                // 800000
    const int* row = ei;
    const int* col = ei + E;

    // workspace carve-up (512B-aligned slices)
    char* ws = (char*)d_ws;
    size_t cur = 0;
    auto take = [&](size_t bytes) { char* p = ws + cur;
                                    cur = (cur + bytes + 511) & ~(size_t)511; return p; };
    float* dinv = (float*)take((size_t)N * 4);
    int*   cnt  = (int*)  take((size_t)N * 4);
    int*   off  = (int*)  take((size_t)N * 4);
    int*   pos  = (int*)  take((size_t)N * 4);
    int*   srcs = (int*)  take((size_t)E * 4);
    float* h1   = (float*)take((size_t)N * NFEAT * 4);
    float* h2   = (float*)take((size_t)N * NFEAT * 4);

    const int TB = 256;
    int nbN = (N + TB - 1) / TB;
    int nbE = (E + TB - 1) / TB;

    // 1) in-degree counts -> dinv, CSR offsets, CSR fill
    k_zero_cnt<<<nbN, TB, 0, stream>>>(cnt, N);
    k_count<<<nbE, TB, 0, stream>>>(col, cnt, E);
    k_dinv<<<nbN, TB, 0, stream>>>(cnt, dinv, N);
    k_scan<<<1, 1024, 0, stream>>>(cnt, off, pos, N);
    k_fill<<<nbE, TB, 0, stream>>>(row, col, pos, srcs, E);

    // 2) two pull hops (atomic-free, L2-resident gathers)
    int nbPull = (N * 32 + TB - 1) / TB;          // one wave32 per node
    k_pull<<<nbPull, TB, 0, stream>>>(x,  h1, off, cnt, srcs, dinv, N);
    k_pull<<<nbPull, TB, 0, stream>>>(h1, h2, off, cnt, srcs, dinv, N);

    // 3) fused WMMA GEMM + bias + log_softmax
    int nRowTiles = N / 16;                       // 3125
    int nBlocks   = (nRowTiles + 4) / 5;          // 625
    k_gemm_logsoftmax<<<nBlocks, 160, 0, stream>>>(h2, W, b, out, nRowTiles);
}